// LossFunctionNcc_Registration_23270132809775
// MI455X (gfx1250) — compile-verified
//
#include <hip/hip_runtime.h>
#include <hip/hip_bf16.h>

typedef __attribute__((ext_vector_type(2))) float v2f;
typedef __attribute__((ext_vector_type(8))) float v8f;

#define NDIM   160
#define WINSZ  9
#define PADR   4          // WIN/2
#define TILE   16
#define HALO   24         // TILE + 2*PADR
#define ZCHUNK 16

// ---------------------------------------------------------------------------
// Kernel 0: zero the two scalar accumulators in workspace (every call).
// ---------------------------------------------------------------------------
__global__ void init_acc_kernel(float* ws) {
    if (threadIdx.x < 2) ws[threadIdx.x] = 0.0f;
}

// ---------------------------------------------------------------------------
// Kernel 1: fused local-NCC.  One wave32 per 16x16 xy tile x 16-z chunk.
// x-sum: 9-tap sliding sum from LDS.  y-sum: v = S(16x24) * U(24x16) via
// six chained V_WMMA_F32_16X16X4_F32.  z-sum: rolling 9-window with LDS ring.
// ---------------------------------------------------------------------------
__launch_bounds__(32)
__global__ void ncc_fused_kernel(const float* __restrict__ I,   // tgt
                                 const float* __restrict__ J,   // y
                                 float* __restrict__ ws_acc) {
    __shared__ float sI[HALO * HALO];
    __shared__ float sJ[HALO * HALO];
    __shared__ float U[5][HALO * TILE];          // x-summed quantities
    __shared__ float ring[5 * 9 * 32 * 8];       // z ring buffer (D-layout)

    const int lane = threadIdx.x;
    const int x0 = blockIdx.x * TILE;
    const int y0 = blockIdx.y * TILE;
    const int z0 = blockIdx.z * ZCHUNK;

    // Zero the ring buffer (window warm-up relies on it).
    for (int t = lane; t < 5 * 9 * 32 * 8; t += 32) ring[t] = 0.0f;
    __syncthreads();

    // A fragments: banded ones matrix S[m][j] = (m <= j <= m+8).
    // 32-bit A 16x4 layout: VGPR0 = {K0 | K2}, VGPR1 = {K1 | K3} per lane half.
    const int m     = lane & 15;
    const int khalf = (lane < 16) ? 0 : 2;
    v2f afrag[6];
#pragma unroll
    for (int c = 0; c < 6; ++c) {
        const int j0 = 4 * c + khalf;
        const int j1 = j0 + 1;
        afrag[c].x = (j0 >= m && j0 <= m + 8) ? 1.0f : 0.0f;
        afrag[c].y = (j1 >= m && j1 <= m + 8) ? 1.0f : 0.0f;
    }

    v8f acc[5];
#pragma unroll
    for (int q = 0; q < 5; ++q) acc[q] = (v8f){};

    float ccsum = 0.0f;
    const int n = lane & 15;  // D/B column index for this lane

    for (int zi = z0 - PADR; zi <= z0 + ZCHUNK + PADR - 1; ++zi) {
        v8f D[5];
        const bool zvalid = (zi >= 0) && (zi < NDIM);
        if (zvalid) {
            const size_t zoff = (size_t)zi * NDIM * NDIM;
            // ---- load 24x24 halo slice of I and J with zero padding ----
            for (int t = lane; t < HALO * HALO; t += 32) {
                const int ly = t / HALO, lx = t % HALO;
                const int gy = y0 - PADR + ly;
                const int gx = x0 - PADR + lx;
                float vi = 0.0f, vj = 0.0f;
                if ((unsigned)gy < NDIM && (unsigned)gx < NDIM) {
                    const size_t g = zoff + (size_t)gy * NDIM + gx;
                    vi = I[g];
                    vj = J[g];
                    if (t < 32 && zi + 1 < NDIM) {    // prefetch next slice
                        __builtin_prefetch(&I[g + (size_t)NDIM * NDIM], 0, 0);
                        __builtin_prefetch(&J[g + (size_t)NDIM * NDIM], 0, 0);
                    }
                }
                sI[t] = vi;
                sJ[t] = vj;
            }
            __syncthreads();

            // ---- x-direction 9-tap sliding sums of the 5 quantities ----
            for (int t = lane; t < HALO * TILE; t += 32) {
                const int y  = t / TILE;
                const int xo = t % TILE;
                float si = 0.f, sj = 0.f, si2 = 0.f, sj2 = 0.f, sij = 0.f;
#pragma unroll
                for (int k = 0; k < WINSZ; ++k) {
                    const float a = sI[y * HALO + xo + k];
                    const float b = sJ[y * HALO + xo + k];
                    si += a; sj += b; si2 += a * a; sj2 += b * b; sij += a * b;
                }
                U[0][t] = si; U[1][t] = sj; U[2][t] = si2; U[3][t] = sj2; U[4][t] = sij;
            }
            __syncthreads();

            // ---- y-direction 9-tap sum as matmul: D = S(16x24) x U(24x16) ----
#pragma unroll
            for (int q = 0; q < 5; ++q) {
                v8f d = (v8f){};
#pragma unroll
                for (int c = 0; c < 6; ++c) {
                    const int j0 = 4 * c + khalf;
                    v2f b;
                    b.x = U[q][j0 * TILE + n];
                    b.y = U[q][(j0 + 1) * TILE + n];
                    d = __builtin_amdgcn_wmma_f32_16x16x4_f32(
                            /*neg_a=*/false, afrag[c],
                            /*neg_b=*/false, b,
                            /*c_mod=*/(short)0, d,
                            /*reuse_a=*/false, /*reuse_b=*/false);
                }
                D[q] = d;
            }
            __syncthreads();
        } else {
#pragma unroll
            for (int q = 0; q < 5; ++q) D[q] = (v8f){};
        }

        // ---- rolling z window: acc += D_zi - D_{zi-9} via LDS ring ----
        const int slot = (zi + 36) % 9;   // zi >= -4, 36 % 9 == 0
#pragma unroll
        for (int q = 0; q < 5; ++q) {
            const int base = ((q * 9 + slot) * 32 + lane) * 8;
#pragma unroll
            for (int r = 0; r < 8; ++r) {
                const float old = ring[base + r];
                ring[base + r] = D[q][r];
                acc[q][r] += D[q][r] - old;
            }
        }

        // ---- emit output slice z_out = zi - 4 ----
        const int zo = zi - PADR;
        if (zo >= z0) {
            const float winsz = (float)(WINSZ * WINSZ * WINSZ);   // 729
            const float inv   = 1.0f / winsz;
#pragma unroll
            for (int r = 0; r < 8; ++r) {
                const float Is  = acc[0][r], Js  = acc[1][r];
                const float I2s = acc[2][r], J2s = acc[3][r], IJs = acc[4][r];
                const float uI = Is * inv, uJ = Js * inv;
                const float cross = IJs - uJ * Is - uI * Js + uI * uJ * winsz;
                const float Ivar  = I2s - 2.0f * uI * Is + uI * uI * winsz;
                const float Jvar  = J2s - 2.0f * uJ * Js + uJ * uJ * winsz;
                ccsum += cross * cross / (Ivar * Jvar + 1e-5f);
            }
        }
    }

    // wave32 reduction + one atomic per block
#pragma unroll
    for (int o = 16; o > 0; o >>= 1) ccsum += __shfl_down(ccsum, o, 32);
    if (lane == 0) atomicAdd(&ws_acc[0], ccsum);
}

// ---------------------------------------------------------------------------
// Kernel 2: flow gradient L2 (sum of squared finite differences, all 3 axes).
// ---------------------------------------------------------------------------
__global__ void grad_l2_kernel(const float* __restrict__ F, float* __restrict__ ws_acc) {
    const long long N = 3LL * NDIM * NDIM * NDIM;
    const long long stride = (long long)gridDim.x * blockDim.x;
    float s = 0.0f;
    for (long long idx = (long long)blockIdx.x * blockDim.x + threadIdx.x;
         idx < N; idx += stride) {
        const int w = (int)(idx % NDIM);
        const long long t1 = idx / NDIM;
        const int h = (int)(t1 % NDIM);
        const long long t2 = t1 / NDIM;
        const int d = (int)(t2 % NDIM);
        const float v = F[idx];
        if (d < NDIM - 1) { const float df = F[idx + NDIM * NDIM] - v; s += df * df; } // dy
        if (h < NDIM - 1) { const float df = F[idx + NDIM] - v;        s += df * df; } // dx
        if (w < NDIM - 1) { const float df = F[idx + 1] - v;           s += df * df; } // dz
    }
#pragma unroll
    for (int o = 16; o > 0; o >>= 1) s += __shfl_down(s, o, 32);
    __shared__ float red[8];
    const int wid = threadIdx.x >> 5, ln = threadIdx.x & 31;
    if (ln == 0) red[wid] = s;
    __syncthreads();
    if (threadIdx.x == 0) {
        float tot = 0.0f;
        const int nw = (int)(blockDim.x >> 5);
        for (int i = 0; i < nw; ++i) tot += red[i];
        atomicAdd(&ws_acc[1], tot);
    }
}

// ---------------------------------------------------------------------------
// Kernel 3: combine scalars -> (loss, ncc, grad)
// ---------------------------------------------------------------------------
__global__ void finalize_kernel(const float* __restrict__ ws_acc, float* __restrict__ out) {
    const float nvox  = (float)NDIM * (float)NDIM * (float)NDIM;
    const float ncc   = -(ws_acc[0] / nvox);
    const float denom = 3.0f * (3.0f * (float)(NDIM - 1) * (float)NDIM * (float)NDIM);
    const float grad  = ws_acc[1] / denom;
    out[0] = 10.0f * ncc + 15.0f * grad;
    out[1] = ncc;
    out[2] = grad;
}

// ---------------------------------------------------------------------------
extern "C" void kernel_launch(void* const* d_in, const int* in_sizes, int n_in,
                              void* d_out, int out_size, void* d_ws, size_t ws_size,
                              hipStream_t stream) {
    const float* y    = (const float*)d_in[0];   // J
    const float* tgt  = (const float*)d_in[1];   // I
    // d_in[2] (src) is unused by the reference.
    const float* flow = (const float*)d_in[3];
    float* ws  = (float*)d_ws;
    float* out = (float*)d_out;

    init_acc_kernel<<<1, 32, 0, stream>>>(ws);

    dim3 grid(NDIM / TILE, NDIM / TILE, NDIM / ZCHUNK);   // 10 x 10 x 10
    ncc_fused_kernel<<<grid, 32, 0, stream>>>(tgt, y, ws);

    grad_l2_kernel<<<2048, 256, 0, stream>>>(flow, ws);

    finalize_kernel<<<1, 1, 0, stream>>>(ws, out);
}